// TGNNPO_89962384982713
// MI455X (gfx1250) — compile-verified
//
#include <hip/hip_runtime.h>

// ---------------- problem constants (from reference) ----------------
#define N_NODES 50000
#define F_IN    128
#define T_P     12
#define E_EDGES 800000
#define MID     32
#define NT      (N_NODES * T_P)   // 600000 rows in the fused GEMM (multiple of 16)
#define YCOLS   64                // [Y_l (32) | Y_r (32)]
#define LDS_PAD 130               // 128 + 2 to break LDS bank conflicts

typedef __attribute__((ext_vector_type(2))) float v2f;
typedef __attribute__((ext_vector_type(8))) float v8f;

// ---------------- deterministic hash-based gumbel -------------------
__device__ __forceinline__ float tg_uniform(unsigned v) {
    v = v * 747796405u + 2891336453u;
    unsigned w = ((v >> ((v >> 28u) + 4u)) ^ v) * 277803737u;
    w = (w >> 22u) ^ w;
    return ((float)(w >> 8) + 0.5f) * (1.0f / 16777216.0f);
}
__device__ __forceinline__ float tg_gumbel(unsigned v) {
    float u = tg_uniform(v);
    return -__logf(-__logf(u));
}

// ---------------- kernel 1: in-degree (exact grid, no guard) --------
__global__ __launch_bounds__(256) void tg_deg(const int* __restrict__ ei,
                                              unsigned* __restrict__ deg) {
    int e = blockIdx.x * 256 + threadIdx.x;
    atomicAdd(&deg[ei[E_EDGES + e]], 1u);
}

// ---------------- kernel 2: fused projection GEMM (f32 WMMA) --------
// Y[r, 0:32]  = x[n,:,t] @ W_l^T   (r = n*T + t)
// Y[r, 32:64] = x[n,:,t] @ W_r^T
// Each wave: 16 rows x 64 cols = four 16x16 C tiles, K looped 128/4.
__global__ __launch_bounds__(256) void tg_gemm(const float* __restrict__ x,
                                               const float* __restrict__ Wl,
                                               const float* __restrict__ Wr,
                                               float* __restrict__ Y) {
    __shared__ float ldsw[YCOLS * LDS_PAD]; // [col][k], padded

    // cooperative stage of W_l / W_r (32 KB of weights, reused by all waves)
    for (int i = threadIdx.x; i < YCOLS * F_IN; i += 256) {
        int col = i >> 7, k = i & 127;
        float w = (col < MID) ? Wl[col * F_IN + k] : Wr[(col - MID) * F_IN + k];
        ldsw[col * LDS_PAD + k] = w;
    }
    __syncthreads();

    const int wave = threadIdx.x >> 5;
    const int lane = threadIdx.x & 31;
    const int half = lane >> 4;   // lanes 16..31 carry K = 2,3 (16x4 f32 layout)
    const int lid  = lane & 15;
    const int rowBase = blockIdx.x * 128 + wave * 16;

    // per-lane A row pointer (clamped so EXEC stays all-ones for WMMA)
    int rA = rowBase + lid;
    rA = (rA < NT) ? rA : (NT - 1);
    const int nA = rA / T_P;
    const int tA = rA - nA * T_P;
    const float* __restrict__ xrow = x + (size_t)nA * (F_IN * T_P) + tA; // stride T_P per feature

    v8f acc0 = {}, acc1 = {}, acc2 = {}, acc3 = {};

    for (int k0 = 0; k0 < F_IN; k0 += 4) {
        const int ka = k0 + 2 * half;
        v2f a;
        a.x = xrow[(size_t)ka * T_P];
        a.y = xrow[(size_t)(ka + 1) * T_P];

        const int c0 = lid;           // tile 0 col
        v2f b0, b1, b2, b3;
        b0.x = ldsw[(c0      ) * LDS_PAD + ka];  b0.y = ldsw[(c0      ) * LDS_PAD + ka + 1];
        b1.x = ldsw[(c0 + 16 ) * LDS_PAD + ka];  b1.y = ldsw[(c0 + 16 ) * LDS_PAD + ka + 1];
        b2.x = ldsw[(c0 + 32 ) * LDS_PAD + ka];  b2.y = ldsw[(c0 + 32 ) * LDS_PAD + ka + 1];
        b3.x = ldsw[(c0 + 48 ) * LDS_PAD + ka];  b3.y = ldsw[(c0 + 48 ) * LDS_PAD + ka + 1];

        acc0 = __builtin_amdgcn_wmma_f32_16x16x4_f32(false, a, false, b0, (short)0, acc0, false, false);
        acc1 = __builtin_amdgcn_wmma_f32_16x16x4_f32(false, a, false, b1, (short)0, acc1, false, false);
        acc2 = __builtin_amdgcn_wmma_f32_16x16x4_f32(false, a, false, b2, (short)0, acc2, false, false);
        acc3 = __builtin_amdgcn_wmma_f32_16x16x4_f32(false, a, false, b3, (short)0, acc3, false, false);
    }

    // NT and rowBase are both multiples of 16, so the 16-row strip is either
    // fully in range or fully out: one wave-uniform branch, unconditional stores.
    if (rowBase < NT) {
        float* yr = Y + (size_t)(rowBase + 8 * half) * YCOLS + lid;
#pragma unroll
        for (int j = 0; j < 8; ++j) {
            yr[(size_t)j * YCOLS + 0]  = acc0[j];
            yr[(size_t)j * YCOLS + 16] = acc1[j];
            yr[(size_t)j * YCOLS + 32] = acc2[j];
            yr[(size_t)j * YCOLS + 48] = acc3[j];
        }
    }
}

// ---------------- kernel 3: edge aggregation (one wave per edge) ----
// agg[dst, t, m] += Y[src*T + t, m]  for m in 0..31, t in 0..11
// Exact grid: blocks * 8 waves == E_EDGES, no tail guard.
__global__ __launch_bounds__(256) void tg_agg(const int* __restrict__ ei,
                                              const float* __restrict__ Y,
                                              float* __restrict__ agg) {
    int warp = (blockIdx.x * 256 + threadIdx.x) >> 5;
    int m = threadIdx.x & 31;
    const int src = ei[warp];
    const int dst = ei[E_EDGES + warp];
    const float* __restrict__ ys = Y + (size_t)src * (T_P * YCOLS);
    float* __restrict__ ad = agg + (size_t)dst * (T_P * MID);
    __builtin_prefetch(ad + m, 0, 0);   // global_prefetch of the atomic target line
#pragma unroll
    for (int t = 0; t < T_P; ++t) {
        float v = ys[t * YCOLS + m];              // coalesced 128B read of Y_l row
        atomicAdd(&ad[t * MID + m], v);           // contiguous f32 atomics (L2-resident)
    }
}

// ---------------- kernel 4: gumbel-softmax of attention (T=12) ------
__global__ void tg_probs(const float* __restrict__ att, float* __restrict__ probs) {
    if (threadIdx.x == 0) {
        float zt[T_P];
        float mx = -3.4e38f;
        for (int t = 0; t < T_P; ++t) {
            zt[t] = att[t] + tg_gumbel(0x13572468u + t);
            mx = fmaxf(mx, zt[t]);
        }
        float s = 0.0f;
        for (int t = 0; t < T_P; ++t) { zt[t] = __expf(zt[t] - mx); s += zt[t]; }
        float inv = 1.0f / s;
        for (int t = 0; t < T_P; ++t) probs[t] = zt[t] * inv;
    }
}

// ---------------- kernel 5: per-node combine (one wave per node) ----
// Exact grid: blocks * 8 waves == N_NODES.
__global__ __launch_bounds__(256) void tg_combine(const float* __restrict__ agg,
                                                  const float* __restrict__ Y,
                                                  const unsigned* __restrict__ deg,
                                                  const float* __restrict__ bl,
                                                  const float* __restrict__ probs,
                                                  const float* __restrict__ linW,
                                                  const float* __restrict__ linb,
                                                  float* __restrict__ z) {
    int n = (blockIdx.x * 256 + threadIdx.x) >> 5;
    int m = threadIdx.x & 31;

    const float invd = 1.0f / fmaxf((float)deg[n], 1.0f);
    const float blm = bl[m];
    float h = 0.0f;
#pragma unroll
    for (int t = 0; t < T_P; ++t) {
        float v = agg[((size_t)n * T_P + t) * MID + m] * invd + blm
                + Y[((size_t)n * T_P + t) * YCOLS + MID + m];   // root term x@W_r^T
        float sq = v * v;
#pragma unroll
        for (int o = 16; o > 0; o >>= 1) sq += __shfl_xor(sq, o, 32);
        float inv_nrm = 1.0f / fmaxf(sqrtf(sq), 1e-12f);
        h += probs[t] * (v * inv_nrm);
    }
    h = (h > 0.0f) ? h : 0.01f * h;                 // leaky_relu
    float s = h * linW[m];
#pragma unroll
    for (int o = 16; o > 0; o >>= 1) s += __shfl_xor(s, o, 32);
    if (m == 0) z[n] = s + linb[0] + tg_gumbel(0xBEEF0000u + (unsigned)n);
}

// ---------------- kernel 6: max + sum-exp reduction (1 block) -------
__global__ __launch_bounds__(1024) void tg_reduce(const float* __restrict__ z,
                                                  float* __restrict__ red) {
    __shared__ float smax[32];
    __shared__ float ssum[32];
    __shared__ float bmax;
    const int tid = threadIdx.x, lane = tid & 31, w = tid >> 5;

    float mx = -3.4e38f;
    for (int i = tid; i < N_NODES; i += 1024) mx = fmaxf(mx, z[i]);
#pragma unroll
    for (int o = 16; o > 0; o >>= 1) mx = fmaxf(mx, __shfl_xor(mx, o, 32));
    if (lane == 0) smax[w] = mx;
    __syncthreads();
    if (w == 0) {
        float v = smax[lane];
#pragma unroll
        for (int o = 16; o > 0; o >>= 1) v = fmaxf(v, __shfl_xor(v, o, 32));
        if (lane == 0) bmax = v;
    }
    __syncthreads();
    const float M = bmax;

    float s = 0.0f;
    for (int i = tid; i < N_NODES; i += 1024) s += __expf(z[i] - M);
#pragma unroll
    for (int o = 16; o > 0; o >>= 1) s += __shfl_xor(s, o, 32);
    if (lane == 0) ssum[w] = s;
    __syncthreads();
    if (w == 0) {
        float v = ssum[lane];
#pragma unroll
        for (int o = 16; o > 0; o >>= 1) v += __shfl_xor(v, o, 32);
        if (lane == 0) { red[0] = M; red[1] = v; }
    }
}

// ---------------- kernel 7: final softmax over nodes ----------------
__global__ __launch_bounds__(256) void tg_softmax(const float* __restrict__ z,
                                                  const float* __restrict__ red,
                                                  float* __restrict__ out) {
    int n = blockIdx.x * 256 + threadIdx.x;
    if (n < N_NODES) out[n] = __expf(z[n] - red[0]) / red[1];
}

// ---------------- host launcher -------------------------------------
extern "C" void kernel_launch(void* const* d_in, const int* in_sizes, int n_in,
                              void* d_out, int out_size, void* d_ws, size_t ws_size,
                              hipStream_t stream) {
    (void)in_sizes; (void)n_in; (void)out_size; (void)ws_size;

    const float* x    = (const float*)d_in[0];   // [1,N,F_IN,T]
    const int*   ei   = (const int*)  d_in[1];   // [2,E]
    // d_in[2] edge_weight: unused by reference
    const float* Wl   = (const float*)d_in[3];   // [MID,F_IN]
    const float* bl   = (const float*)d_in[4];   // [MID]
    const float* Wr   = (const float*)d_in[5];   // [MID,F_IN]
    const float* att  = (const float*)d_in[6];   // [T]
    const float* linW = (const float*)d_in[7];   // [1,MID]
    const float* linb = (const float*)d_in[8];   // [1]
    float* out = (float*)d_out;                  // [N,1]

    // workspace carve-up (256B aligned chunks)
    char* ws = (char*)d_ws;
    size_t off = 0;
    float*    Y     = (float*)(ws + off); off += (size_t)NT * YCOLS * sizeof(float);  // 153.6 MB
    float*    agg   = (float*)(ws + off); off += (size_t)NT * MID * sizeof(float);    //  76.8 MB
    unsigned* deg   = (unsigned*)(ws + off); off += ((size_t)N_NODES * 4 + 255) & ~(size_t)255;
    float*    probs = (float*)(ws + off); off += 256;
    float*    z     = (float*)(ws + off); off += ((size_t)N_NODES * 4 + 255) & ~(size_t)255;
    float*    red   = (float*)(ws + off); off += 256;

    hipMemsetAsync(agg, 0, (size_t)NT * MID * sizeof(float), stream);
    hipMemsetAsync(deg, 0, (size_t)N_NODES * sizeof(unsigned), stream);

    tg_deg    <<<E_EDGES / 256, 256, 0, stream>>>(ei, deg);                 // 3125 blocks, exact
    tg_gemm   <<<(NT + 127) / 128, 256, 0, stream>>>(x, Wl, Wr, Y);         // 4688 blocks
    tg_agg    <<<E_EDGES / 8, 256, 0, stream>>>(ei, Y, agg);                // 100000 blocks, exact
    tg_probs  <<<1, 32, 0, stream>>>(att, probs);
    tg_combine<<<N_NODES / 8, 256, 0, stream>>>(agg, Y, deg, bl, probs, linW, linb, z); // 6250, exact
    tg_reduce <<<1, 1024, 0, stream>>>(z, red);
    tg_softmax<<<(N_NODES + 255) / 256, 256, 0, stream>>>(z, red, out);
}